// VectorQuantizer_86242943304022
// MI455X (gfx1250) — compile-verified
//
#include <hip/hip_runtime.h>

// Problem constants (from reference setup_inputs)
#define B_  2
#define D_  2048
#define S_  16
#define N_  256          // S*S
#define H_  256
#define K_  4096
#define d_  8
#define T_  (K_ / 16)    // 256 code tiles per head

typedef float v2f __attribute__((ext_vector_type(2)));
typedef float v8f __attribute__((ext_vector_type(8)));
typedef unsigned int v4u __attribute__((ext_vector_type(4)));
typedef int v4i_ __attribute__((ext_vector_type(4)));
typedef int v8i_ __attribute__((ext_vector_type(8)));

// D = c_sq + (-2x).c == distance; 8 independent per-slot argmin chains.
#define VQ_STEP(A0, A1, CS, MIDX)                                             \
    do {                                                                      \
        v8f c_ = (CS);                                                        \
        c_ = __builtin_amdgcn_wmma_f32_16x16x4_f32(                           \
                 false, (A0), false, bq0, (short)0, c_, false, false);        \
        c_ = __builtin_amdgcn_wmma_f32_16x16x4_f32(                           \
                 false, (A1), false, bq1, (short)0, c_, false, false);        \
        _Pragma("unroll")                                                     \
        for (int v = 0; v < 8; ++v) {                                         \
            bool lt_ = c_[v] < bestv[v];                                      \
            bestv[v] = lt_ ? c_[v] : bestv[v];                                \
            bm[v]    = lt_ ? (MIDX) : bm[v];                                  \
        }                                                                     \
    } while (0)

// One workgroup per head h. 512 threads = 16 waves (wave32).
// LDS: codebook[h] (128KB) + c_sq (16KB) + 256B pipeline-overread pad.
__launch_bounds__(512)
__global__ void vq_wmma_kernel(const float* __restrict__ emb,
                               const float* __restrict__ cb,
                               float* __restrict__ out) {
    extern __shared__ float lds[];
    float* lds_cb  = lds;             // K_*d_ floats (128KB)
    float* lds_csq = lds + K_ * d_;   // K_ floats (16KB) + pad after

    const int h   = blockIdx.x;
    const int tid = threadIdx.x;
    const float* cbh = cb + (size_t)h * K_ * d_;

    // ---- Stage codebook[h] (128KB) into LDS via the Tensor Data Mover ----
    // One TDM descriptor: 2D tensor, data_size=4B, 8192 elems/row x 4 rows,
    // tile == tensor -> single instruction DMAs all 128KB, no VGPR traffic.
    if (tid < 32) {   // uniform: wave 0 only issues the TDM op
        const unsigned long long ga = (unsigned long long)(uintptr_t)cbh;
        const unsigned ldsaddr = (unsigned)(uintptr_t)lds_cb; // LDS byte offset

        v4u g0;
        g0.x = 1u;                                      // count=1 (user D#)
        g0.y = ldsaddr;                                 // lds_addr
        g0.z = (unsigned)ga;                            // global_addr[31:0]
        g0.w = ((unsigned)(ga >> 32) & 0x01FFFFFFu)     // global_addr[56:32]
               | (2u << 30);                            // type=2 ("image")

        v8i_ g1;
        g1[0] = (int)(2u << 16);                        // wg_mask=0, data_size=4B
        g1[1] = (int)(8192u << 16);                     // tensor_dim0 = 8192 (lo16)
        g1[2] = (int)(4u << 16);                        // dim0 hi=0 | tensor_dim1=4
        g1[3] = (int)(8192u << 16);                     // dim1 hi=0 | tile_dim0=8192
        g1[4] = (int)4u;                                // tile_dim1=4 | tile_dim2=0
        g1[5] = (int)8192u;                             // tensor_dim0_stride lo32
        g1[6] = 0;                                      // stride0 hi | stride1 lo
        g1[7] = 0;                                      // stride1 hi

        v4i_ g2; g2[0] = 1; g2[1] = 1; g2[2] = 0; g2[3] = 0;   // dim2=1, dim3=1
        v4i_ g3; g3[0] = 0; g3[1] = (int)(1u << 16); g3[2] = 0; g3[3] = 0; // dim4=1

#if defined(__clang_major__) && (__clang_major__ >= 23)
        v8i_ g4; g4[0]=0; g4[1]=0; g4[2]=0; g4[3]=0; g4[4]=0; g4[5]=0; g4[6]=0; g4[7]=0;
        __builtin_amdgcn_tensor_load_to_lds(g0, g1, g2, g3, g4, 0);
#else
        __builtin_amdgcn_tensor_load_to_lds(g0, g1, g2, g3, 0);
#endif
        __builtin_amdgcn_s_wait_tensorcnt(0);
    }
    __syncthreads();

    // ---- Precompute c_sq = ||c_k||^2 into LDS ----
    for (int k = tid; k < K_; k += 512) {
        const float* c = lds_cb + k * d_;
        float s = 0.f;
        #pragma unroll
        for (int j = 0; j < d_; ++j) s += c[j] * c[j];
        lds_csq[k] = s;
    }
    __syncthreads();

    const int wave = tid >> 5;      // 0..15
    const int lane = tid & 31;
    const int half = lane >> 4;     // 0 or 1 (upper lane half)
    const int ln   = lane & 15;     // query column within tile

    // Each wave handles 2 tiles of 16 queries (32 tiles of 512 queries total).
    for (int pass = 0; pass < 2; ++pass) {
        const int tile = wave + pass * 16;      // 0..31
        const int b    = tile >> 4;             // batch
        const int n    = ((tile & 15) << 4) + ln;

        // Query B-matrix (4x16 per k-step): VGPR0=K(0+2*half), VGPR1=K(1+2*half),
        // N=lane%16. x[n][k] = emb[((b*D + h*8 + k)*N_) + n]; coalesced over n.
        const float* ebase = emb + ((size_t)b * D_ + h * d_) * N_ + n;
        v2f bq0, bq1;
        bq0.x = -2.0f * ebase[(0 + 2 * half) * N_];
        bq0.y = -2.0f * ebase[(1 + 2 * half) * N_];
        bq1.x = -2.0f * ebase[(4 + 2 * half) * N_];
        bq1.y = -2.0f * ebase[(5 + 2 * half) * N_];

        float bestv[8];
        int   bm[8];
        #pragma unroll
        for (int v = 0; v < 8; ++v) { bestv[v] = 3.4028235e38f; bm[v] = 0; }

        // A-tile (codes x k), 32-bit A 16x4 layout: VGPR0=K(0+2*half),
        // VGPR1=K(1+2*half), M=lane%16. One tile = 128 floats of codebook,
        // 16 floats of c_sq. Software-pipelined: loads for tile m+1 issue
        // before the WMMAs/selects of tile m (ping-pong register sets).
        const float* abase = lds_cb  + ln * d_ + 2 * half;
        const float* cbase = lds_csq + 8 * half;

        v2f a0A = *(const v2f*)(abase);
        v2f a1A = *(const v2f*)(abase + 4);
        v8f csA = *(const v8f*)(cbase);
        v2f a0B, a1B;
        v8f csB;

        for (int m = 0; m < T_; m += 2) {
            a0B = *(const v2f*)(abase + (m + 1) * 128);
            a1B = *(const v2f*)(abase + (m + 1) * 128 + 4);
            csB = *(const v8f*)(cbase + (m + 1) * 16);
            VQ_STEP(a0A, a1A, csA, m);

            a0A = *(const v2f*)(abase + (m + 2) * 128);   // over-reads 1 tile at
            a1A = *(const v2f*)(abase + (m + 2) * 128 + 4); // end (padded LDS)
            csA = *(const v8f*)(cbase + (m + 2) * 16);
            VQ_STEP(a0B, a1B, csB, m + 1);
        }

        // Merge the 8 slots; tie -> lowest full code index (jnp.argmin order).
        // Branchless: non-short-circuit ops -> v_cndmask, no exec juggling.
        float best = bestv[0];
        int   bidx = bm[0] * 16 + 8 * half;
        #pragma unroll
        for (int v = 1; v < 8; ++v) {
            int  code   = bm[v] * 16 + 8 * half + v;
            bool better = (bestv[v] < best) | ((bestv[v] == best) & (code < bidx));
            best = better ? bestv[v] : best;
            bidx = better ? code : bidx;
        }

        // Merge the two lane-halves (same query n, disjoint code halves).
        {
            float obest  = __shfl_xor(best, 16, 32);
            int   oidx   = __shfl_xor(bidx, 16, 32);
            bool  better = (obest < best) | ((obest == best) & (oidx < bidx));
            best = better ? obest : best;
            bidx = better ? oidx : bidx;
        }

        // Emit quantized vector: out[b][h*8+j][n] = cb[h][bidx][j]
        if (half == 0) {
            float* obase = out + ((size_t)b * D_ + h * d_) * N_ + n;
            const float* q = lds_cb + bidx * d_;
            #pragma unroll
            for (int j = 0; j < d_; ++j) obase[j * N_] = q[j];
        }
    }
}

extern "C" void kernel_launch(void* const* d_in, const int* in_sizes, int n_in,
                              void* d_out, int out_size, void* d_ws, size_t ws_size,
                              hipStream_t stream) {
    const float* emb = (const float*)d_in[0];   // (B, D, S, S) f32
    const float* cb  = (const float*)d_in[1];   // (H, K, d)    f32
    float* out = (float*)d_out;                 // (B, D, S, S) f32

    // 128KB codebook + 16KB c_sq + 256B pad for pipeline over-read = 147712B.
    const size_t shmem = (size_t)(K_ * d_ + K_ + 64) * sizeof(float);
    (void)hipFuncSetAttribute((const void*)vq_wmma_kernel,
                              hipFuncAttributeMaxDynamicSharedMemorySize,
                              (int)shmem);
    vq_wmma_kernel<<<H_, 512, shmem, stream>>>(emb, cb, out);
    (void)in_sizes; (void)n_in; (void)out_size; (void)d_ws; (void)ws_size;
}